// DigitCaps_7009386627666
// MI455X (gfx1250) — compile-verified
//
#include <hip/hip_runtime.h>

// DigitCaps dynamic routing, fused: x_hat is never materialized (378 MB in the
// reference). Each workgroup owns one (digit-capsule j, 16-batch tile); the
// routing logits b[16][1152] live in LDS; x (18.9 MB) and W (5.9 MB) stream
// from L2 every sweep. All contractions run on V_WMMA_F32_16X16X4_F32.

typedef float v2f __attribute__((ext_vector_type(2)));
typedef float v8f __attribute__((ext_vector_type(8)));

#define CAPS_B 512
#define CAPS_I 1152
#define CAPS_D 8
#define CAPS_J 10
#define CAPS_C 16
// Row stride for b-logits in LDS. 1156 mod 64 = 4, so the 16 rows accessed by
// lanes m=0..15 map to banks (4m+i) mod 64 -> 16 distinct banks: conflict-free.
// (1152 would be 16-way conflicted; 1160 would still be 2-way: 8*8=64.)
#define IPAD   1156
#define NWAVE  4
#define IW     (CAPS_I / NWAVE)   // 288 i's per wave
#define IHALF  (IW / 2)           // 144 per half-wave (softmax stat passes)

// WMMA f32 16x16x4 register layout assumptions (ISA 7.12.2):
//  A (16x4): lane l holds rows m=l&15; VGPR v holds K = v + 2*(l>>4)
//  B (4x16): lane l holds cols n=l&15; VGPR v holds K = v + 2*(l>>4)  (mirror of A)
//  C/D (16x16): lane l holds col n=l&15; VGPR v holds row M = v + 8*(l>>4)

__global__ __launch_bounds__(128) void digitcaps_route(
    const float* __restrict__ x,   // (B, I, D)
    const float* __restrict__ W,   // (J, I, D, C)
    float* __restrict__ out)       // (B, J, C)
{
  extern __shared__ float lds[];
  float* s_b   = lds;                  // [16][IPAD] routing logits for this tile
  float* s_red = lds + 16 * IPAD;      // [NWAVE][16] per-row softmax partials
  float* s_acc = s_red + NWAVE * 16;   // [NWAVE][256] per-wave s-tile partials

  const int lane = threadIdx.x & 31;
  const int wv   = threadIdx.x >> 5;
  const int m    = lane & 15;          // A row (batch) / B,D column (channel)
  const int half = lane >> 4;

  const int j  = blockIdx.x % CAPS_J;
  const int b0 = (blockIdx.x / CAPS_J) * 16;

  // b logits start at zero
  for (int idx = threadIdx.x; idx < 16 * IPAD; idx += blockDim.x) s_b[idx] = 0.0f;
  __syncthreads();

  const int i0 = wv * IW;
  const int i1 = i0 + IW;

  const float* xb = x + (size_t)(b0 + m) * (CAPS_I * CAPS_D);          // x[b0+m, i, d]
  const float* Wj = W + (size_t)j * (CAPS_I * CAPS_D * CAPS_C);        // W[j, i, d, c]

  v8f vdig = {0.f, 0.f, 0.f, 0.f, 0.f, 0.f, 0.f, 0.f};

  for (int iter = 0; iter < 3; ++iter) {
    // ---- softmax stats over I for each of the 16 batch rows ----
    float pmax = -1e30f;
    for (int i = i0 + half * IHALF; i < i0 + (half + 1) * IHALF; ++i)
      pmax = fmaxf(pmax, s_b[m * IPAD + i]);
    pmax = fmaxf(pmax, __shfl_xor(pmax, 16, 32));
    if (half == 0) s_red[wv * 16 + m] = pmax;
    __syncthreads();
    const float rmax = fmaxf(fmaxf(s_red[m], s_red[16 + m]),
                             fmaxf(s_red[32 + m], s_red[48 + m]));
    __syncthreads();
    float psum = 0.f;
    for (int i = i0 + half * IHALF; i < i0 + (half + 1) * IHALF; ++i)
      psum += __expf(s_b[m * IPAD + i] - rmax);
    psum += __shfl_xor(psum, 16, 32);
    if (half == 0) s_red[wv * 16 + m] = psum;
    __syncthreads();
    const float rinv = 1.f / (s_red[m] + s_red[16 + m] + s_red[32 + m] + s_red[48 + m]);
    __syncthreads();

    // ---- s sweep: s[b,c] = sum_{i,d} (c_bi * x[b,i,d]) * W[j,i,d,c] ----
    // GEMM M=16, N=16, K folded = I*D; softmax weight folded into A on the fly.
    v8f acc = {0.f, 0.f, 0.f, 0.f, 0.f, 0.f, 0.f, 0.f};
    for (int i = i0; i < i1; ++i) {
      const float cmi = __expf(s_b[m * IPAD + i] - rmax) * rinv;
      const float* xp = xb + i * CAPS_D + 2 * half;
      v2f a1; a1.x = cmi * xp[0]; a1.y = cmi * xp[1];   // K = 0..3
      v2f a2; a2.x = cmi * xp[4]; a2.y = cmi * xp[5];   // K = 4..7
      const float* wp = Wj + i * (CAPS_D * CAPS_C) + m;
      v2f b1; b1.x = wp[(2 * half + 0) * 16]; b1.y = wp[(2 * half + 1) * 16];
      v2f b2; b2.x = wp[(2 * half + 4) * 16]; b2.y = wp[(2 * half + 5) * 16];
      acc = __builtin_amdgcn_wmma_f32_16x16x4_f32(false, a1, false, b1,
                                                  (short)0, acc, false, false);
      acc = __builtin_amdgcn_wmma_f32_16x16x4_f32(false, a2, false, b2,
                                                  (short)0, acc, false, false);
    }
    // cross-wave reduction of the partial s tiles
#pragma unroll
    for (int v = 0; v < 8; ++v) s_acc[wv * 256 + v * 32 + lane] = acc[v];
    __syncthreads();
#pragma unroll
    for (int v = 0; v < 8; ++v)
      acc[v] = s_acc[v * 32 + lane] + s_acc[256 + v * 32 + lane] +
               s_acc[512 + v * 32 + lane] + s_acc[768 + v * 32 + lane];
    __syncthreads();

    // ---- squash per batch row (row = v + 8*half, cols across 16 lanes) ----
#pragma unroll
    for (int v = 0; v < 8; ++v) {
      const float sv = acc[v];
      float sq = sv * sv;
      sq += __shfl_xor(sq, 8, 16);
      sq += __shfl_xor(sq, 4, 16);
      sq += __shfl_xor(sq, 2, 16);
      sq += __shfl_xor(sq, 1, 16);
      const float nrm = sqrtf(sq);
      const float scale = (sq / (1.f + sq)) / (nrm + 1e-7f);
      vdig[v] = sv * scale;                 // same C/D layout as x_hat tiles
    }

    if (iter < 2) {
      // ---- agreement sweep: b[b,i] += sum_c x_hat[b,i,c] * v[b,c] ----
      for (int i = i0; i < i1; ++i) {
        const float* xp = xb + i * CAPS_D + 2 * half;
        v2f a1; a1.x = xp[0]; a1.y = xp[1];
        v2f a2; a2.x = xp[4]; a2.y = xp[5];
        const float* wp = Wj + i * (CAPS_D * CAPS_C) + m;
        v2f b1; b1.x = wp[(2 * half + 0) * 16]; b1.y = wp[(2 * half + 1) * 16];
        v2f b2; b2.x = wp[(2 * half + 4) * 16]; b2.y = wp[(2 * half + 5) * 16];
        v8f t = {0.f, 0.f, 0.f, 0.f, 0.f, 0.f, 0.f, 0.f};
        t = __builtin_amdgcn_wmma_f32_16x16x4_f32(false, a1, false, b1,
                                                  (short)0, t, false, false);
        t = __builtin_amdgcn_wmma_f32_16x16x4_f32(false, a2, false, b2,
                                                  (short)0, t, false, false);
        float ag[8];
#pragma unroll
        for (int v = 0; v < 8; ++v) {
          float p = t[v] * vdig[v];          // identical register layouts
          p += __shfl_xor(p, 8, 16);
          p += __shfl_xor(p, 4, 16);
          p += __shfl_xor(p, 2, 16);
          p += __shfl_xor(p, 1, 16);
          ag[v] = p;                         // row (v + 8*half) agreement, all lanes
        }
        float outv = ag[0];
#pragma unroll
        for (int v = 1; v < 8; ++v)
          if (m == v) outv = ag[v];          // cndmask chain: lane m<8 gets row m
        if (m < 8) s_b[(m + 8 * half) * IPAD + i] += outv;
      }
      __syncthreads();
    }
  }

  // ---- write v: rows v+8*half -> batch b0+row, column m -> channel ----
  if (wv == 0) {
#pragma unroll
    for (int v = 0; v < 8; ++v) {
      const int b = b0 + v + 8 * half;
      out[((size_t)b * CAPS_J + j) * CAPS_C + m] = vdig[v];
    }
  }
}

extern "C" void kernel_launch(void* const* d_in, const int* in_sizes, int n_in,
                              void* d_out, int out_size, void* d_ws, size_t ws_size,
                              hipStream_t stream) {
  const float* x = (const float*)d_in[0];   // (512, 1152, 8)
  const float* W = (const float*)d_in[1];   // (10, 1152, 8, 16)
  float* out = (float*)d_out;               // (512, 10, 16)
  const size_t lds_bytes = (size_t)(16 * IPAD + NWAVE * 16 + NWAVE * 256) * sizeof(float);
  dim3 grid((CAPS_B / 16) * CAPS_J);        // 320 workgroups
  dim3 block(32 * NWAVE);                   // 4 waves, I axis split across waves
  hipLaunchKernelGGL(digitcaps_route, grid, block, lds_bytes, stream, x, W, out);
}